// ANFIS_15410342658136
// MI455X (gfx1250) — compile-verified
//
#include <hip/hip_runtime.h>

typedef float v2f __attribute__((ext_vector_type(2)));
typedef float v8f __attribute__((ext_vector_type(8)));

#define B_ROWS  8192
#define IN_DIM  32
#define HID     256
#define OUT_DIM 256
#define N_RULES 16

// LDS strides chosen for conflict-free ds_load_b64 A-fragment reads:
// stride mod 64 == 4 floats -> consecutive rows land 4 banks apart, 16 rows
// span all 64 banks; the two K-halves (lanes 0-15 vs 16-31) occupy the
// interleaved bank pairs.
#define XS_STRIDE 36    // 32 + 4 pad
#define HS_STRIDE 260   // 256 + 4 pad

__global__ __launch_bounds__(256, 1)
void anfis_fused_kernel(const float* __restrict__ x,
                        const float* __restrict__ W1, const float* __restrict__ b1,
                        const float* __restrict__ W2, const float* __restrict__ b2,
                        const float* __restrict__ W3, const float* __restrict__ b3,
                        const float* __restrict__ centers,
                        const float* __restrict__ widths,
                        const float* __restrict__ params,
                        float* __restrict__ out)
{
    __shared__ float xs[16 * XS_STRIDE];   // x stripe   [16 x 32]
    __shared__ float hs[16 * HS_STRIDE];   // h1 then h2 [16 x 256]
    __shared__ float pbuf[16];             // per-row product of x

    const int tid  = threadIdx.x;
    const int lane = tid & 31;
    const int wave = tid >> 5;
    const int m0   = blockIdx.x * 16;      // 512 blocks * 16 rows = 8192

    // ---- stage x stripe into LDS ----
    for (int i = tid; i < 16 * IN_DIM; i += 256) {
        int r = i >> 5, c = i & 31;
        xs[r * XS_STRIDE + c] = x[(m0 + r) * IN_DIM + c];
    }
    __syncthreads();

    // per-row feature product (out[b,o] = prod_f x[b,f] * w^F)
    if (tid < 16) {
        float p = 1.0f;
        #pragma unroll
        for (int c = 0; c < IN_DIM; ++c) p *= xs[tid * XS_STRIDE + c];
        pbuf[tid] = p;
    }

    // fragment lane coordinates (ISA 7.12.2 layouts, wave32)
    const int arow  = lane & 15;          // A-frag: M = lane%16
    const int khalf = (lane >> 4) * 2;    // A/B-frag: K offset 0 or 2
    const int mhalf = (lane >> 4) * 8;    // C/D-frag: row offset 0 or 8
    const int ncol  = lane & 15;          // C/D/B-frag: N = lane%16

    v8f acc[2];

    // ================= GEMM1: h1 = relu(x @ W1 + b1), K = 32 =================
    #pragma unroll
    for (int t = 0; t < 2; ++t) {
        const int n = (wave * 2 + t) * 16 + ncol;
        v8f c = {};
        #pragma unroll
        for (int k0 = 0; k0 < IN_DIM; k0 += 4) {
            v2f a = *(const v2f*)&xs[arow * XS_STRIDE + k0 + khalf];
            v2f b;
            b.x = W1[(k0 + khalf + 0) * HID + n];
            b.y = W1[(k0 + khalf + 1) * HID + n];
            c = __builtin_amdgcn_wmma_f32_16x16x4_f32(
                    false, a, false, b, (short)0, c, false, false);
        }
        acc[t] = c;
    }
    // bias + relu, store h1 to LDS
    #pragma unroll
    for (int t = 0; t < 2; ++t) {
        const int n = (wave * 2 + t) * 16 + ncol;
        const float bias = b1[n];
        #pragma unroll
        for (int i = 0; i < 8; ++i) {
            float v = acc[t][i] + bias;
            hs[(i + mhalf) * HS_STRIDE + n] = v > 0.0f ? v : 0.0f;
        }
    }
    __syncthreads();

    // ================= GEMM2: h2 = relu(h1 @ W2 + b2), K = 256 ===============
    #pragma unroll
    for (int t = 0; t < 2; ++t) {
        const int n = (wave * 2 + t) * 16 + ncol;
        v8f c = {};
        #pragma unroll 4
        for (int k0 = 0; k0 < HID; k0 += 4) {
            v2f a = *(const v2f*)&hs[arow * HS_STRIDE + k0 + khalf];
            v2f b;
            b.x = W2[(k0 + khalf + 0) * HID + n];
            b.y = W2[(k0 + khalf + 1) * HID + n];
            c = __builtin_amdgcn_wmma_f32_16x16x4_f32(
                    false, a, false, b, (short)0, c, false, false);
        }
        acc[t] = c;
    }
    __syncthreads();   // all waves done reading h1 before overwriting
    #pragma unroll
    for (int t = 0; t < 2; ++t) {
        const int n = (wave * 2 + t) * 16 + ncol;
        const float bias = b2[n];
        #pragma unroll
        for (int i = 0; i < 8; ++i) {
            float v = acc[t][i] + bias;
            hs[(i + mhalf) * HS_STRIDE + n] = v > 0.0f ? v : 0.0f;
        }
    }
    __syncthreads();

    // ================= GEMM3: y = h2 @ W3 + b3, K = 256 ======================
    #pragma unroll
    for (int t = 0; t < 2; ++t) {
        const int n = (wave * 2 + t) * 16 + ncol;
        v8f c = {};
        #pragma unroll 4
        for (int k0 = 0; k0 < HID; k0 += 4) {
            v2f a = *(const v2f*)&hs[arow * HS_STRIDE + k0 + khalf];
            v2f b;
            b.x = W3[(k0 + khalf + 0) * HID + n];
            b.y = W3[(k0 + khalf + 1) * HID + n];
            c = __builtin_amdgcn_wmma_f32_16x16x4_f32(
                    false, a, false, b, (short)0, c, false, false);
        }
        acc[t] = c;
    }

    // ============ fuzzy epilogue: membership, defuzz, ^32, product ===========
    #pragma unroll
    for (int t = 0; t < 2; ++t) {
        const int o = (wave * 2 + t) * 16 + ncol;     // output column
        const float b3v = b3[o];
        const float* __restrict__ cr = &centers[o * N_RULES];
        const float* __restrict__ wr = &widths [o * N_RULES];
        const float* __restrict__ pr = &params [o * N_RULES];
        #pragma unroll
        for (int i = 0; i < 8; ++i) {
            const float y = acc[t][i] + b3v;
            float msum = 0.0f, wsum = 0.0f;
            #pragma unroll
            for (int r = 0; r < N_RULES; ++r) {
                const float z  = (y - cr[r]) / wr[r];
                const float mz = __expf(-z * z);
                msum += mz;
                wsum += mz * pr[r];
            }
            float w = wsum / msum;
            // w^IN_DIM = w^32 via 5 squarings
            w = w * w; w = w * w; w = w * w; w = w * w; w = w * w;
            const int m = i + mhalf;                   // local row 0..15
            out[(m0 + m) * OUT_DIM + o] = pbuf[m] * w;
        }
    }
}

extern "C" void kernel_launch(void* const* d_in, const int* in_sizes, int n_in,
                              void* d_out, int out_size, void* d_ws, size_t ws_size,
                              hipStream_t stream) {
    const float* x       = (const float*)d_in[0];
    const float* W1      = (const float*)d_in[1];
    const float* b1      = (const float*)d_in[2];
    const float* W2      = (const float*)d_in[3];
    const float* b2      = (const float*)d_in[4];
    const float* W3      = (const float*)d_in[5];
    const float* b3      = (const float*)d_in[6];
    const float* centers = (const float*)d_in[7];
    const float* widths  = (const float*)d_in[8];
    const float* params  = (const float*)d_in[9];
    float* out = (float*)d_out;

    dim3 grid(B_ROWS / 16);   // 512 workgroups, one 16-row stripe each
    dim3 block(256);          // 8 wave32 -> 16 N-tiles of 16 cols = 256 cols
    anfis_fused_kernel<<<grid, block, 0, stream>>>(
        x, W1, b1, W2, b2, W3, b3, centers, widths, params, out);
}